// mdpn_Model_modify_755914244693
// MI455X (gfx1250) — compile-verified
//
#include <hip/hip_runtime.h>
#include <math.h>

#define M_FILT 16
#define IMG 96
#define BATCH 16
#define S_OUT 88
#define SP 22
#define FIN (M_FILT * SP * SP)   // 7744
#define H1 128
#define CLASSES 10

typedef __attribute__((ext_vector_type(2))) float v2f;
typedef __attribute__((ext_vector_type(8))) float v8f;

// ---------------------------------------------------------------------------
// Stage 1: dendrite (sum of logs form) + fused 4x4 max-pool.
// grid = BATCH*SP blocks, block = SP*M_FILT = 352 threads (11 wave32s).
// Thread t: m = t%16, sp_c = t/16. Block: b = bid/SP, sp_r = bid%SP.
// Writes pooled activations in [B, SP, SP, M] order (matches torch flatten).
// ---------------------------------------------------------------------------
__global__ __launch_bounds__(352) void dendrite_pool_kernel(
    const float* __restrict__ x,   // [B, 96, 96]
    const float* __restrict__ w,   // [16, 9, 9]
    const float* __restrict__ q,   // [16, 9, 9]
    float* __restrict__ y)         // [B, SP, SP, M]
{
    __shared__ float xs[12 * IMG];          // 12 input rows of this band
    __shared__ float ws[81 * M_FILT];       // 10*w, transposed [rc][m]
    __shared__ float qs[81 * M_FILT];       // 10*q, transposed [rc][m]

    const int bid  = blockIdx.x;
    const int b    = bid / SP;
    const int sp_r = bid % SP;
    const int tid  = threadIdx.x;

    const float* xb = x + b * IMG * IMG + (4 * sp_r) * IMG;
    for (int i = tid; i < 12 * IMG; i += 352) xs[i] = xb[i];
    for (int i = tid; i < 81 * M_FILT; i += 352) {
        int m  = i & 15;
        int rc = i >> 4;
        ws[rc * M_FILT + m] = 10.0f * w[m * 81 + rc];
        qs[rc * M_FILT + m] = 10.0f * q[m * 81 + rc];
    }
    __syncthreads();

    const int m    = tid & 15;
    const int sp_c = tid >> 4;
    const int col0 = 4 * sp_c;

    const float C22PI  = 6.9115038378975450f;  // 2.2 * pi
    const float LOG2PI = 1.8378770664093453f;  // log(2*pi)

    float best = -3.0e38f;
    for (int i = 0; i < 4; ++i) {
        for (int j = 0; j < 4; ++j) {
            float acc = 0.0f;
            #pragma unroll
            for (int r = 0; r < 9; ++r) {
                const float* xrow = &xs[(i + r) * IMG + col0 + j];
                #pragma unroll
                for (int c = 0; c < 9; ++c) {
                    const int rc = r * 9 + c;
                    // u = 10*(x*w - q);  log(t+0.6) = log(2.2pi + 2*atan(u)) - log(2pi)
                    float u = fmaf(xrow[c], ws[rc * M_FILT + m], -qs[rc * M_FILT + m]);
                    acc += __logf(C22PI + 2.0f * atanf(u));
                }
            }
            best = fmaxf(best, acc);
        }
    }
    y[((b * SP + sp_r) * SP + sp_c) * M_FILT + m] = best - 81.0f * LOG2PI;
}

// ---------------------------------------------------------------------------
// Stage 2: H = relu(Y[16,FIN] @ W1^T + b1) via V_WMMA_F32_16X16X4_F32.
// grid = 8 blocks (one N-tile of 16 each), block = 32 threads (one wave).
// A (16x4 f32): VGPR0 lanes0-15 K=0 / lanes16-31 K=2; VGPR1 K=1/K=3.
// B (4x16 f32): symmetric over K; N = lane&15.
// C/D (16x16 f32): vgpr v -> M = v + 8*(lane>=16), N = lane&15.
// ---------------------------------------------------------------------------
__global__ __launch_bounds__(32) void fc1_wmma_kernel(
    const float* __restrict__ Y,    // [16, FIN]
    const float* __restrict__ W1,   // [128, FIN] row-major
    const float* __restrict__ b1,   // [128]
    float* __restrict__ H)          // [16, 128]
{
    const int lane  = threadIdx.x & 31;
    const int ntile = blockIdx.x;        // 0..7
    const int row   = lane & 15;         // A row (M) and B col (N)
    const int hi    = lane >> 4;         // K sub-offset selector

    const float* arow = Y  + row * FIN + 2 * hi;
    const float* brow = W1 + (ntile * 16 + row) * FIN + 2 * hi;

    v8f acc = {};
    #pragma unroll 4
    for (int k0 = 0; k0 < FIN; k0 += 4) {
        v2f a, bf;
        a.x  = arow[k0];
        a.y  = arow[k0 + 1];
        bf.x = brow[k0];
        bf.y = brow[k0 + 1];
        acc = __builtin_amdgcn_wmma_f32_16x16x4_f32(
            /*neg_a=*/false, a, /*neg_b=*/false, bf,
            /*c_mod=*/(short)0, acc, /*reuse_a=*/false, /*reuse_b=*/false);
    }

    const int ncol = ntile * 16 + row;
    const float bias = b1[ncol];
    #pragma unroll
    for (int v = 0; v < 8; ++v) {
        const int mrow = v + 8 * hi;
        H[mrow * H1 + ncol] = fmaxf(acc[v] + bias, 0.0f);
    }
}

// ---------------------------------------------------------------------------
// Stage 3: out = H[16,128] @ W2^T + b2  (tiny; plain VALU dot products)
// ---------------------------------------------------------------------------
__global__ __launch_bounds__(160) void fc2_kernel(
    const float* __restrict__ H,    // [16, 128]
    const float* __restrict__ W2,   // [10, 128]
    const float* __restrict__ b2,   // [10]
    float* __restrict__ out)        // [16, 10]
{
    const int t = threadIdx.x;
    if (t >= BATCH * CLASSES) return;
    const int b = t / CLASSES;
    const int c = t % CLASSES;
    const float* h  = H  + b * H1;
    const float* wv = W2 + c * H1;
    float s = b2[c];
    #pragma unroll 8
    for (int i = 0; i < H1; ++i) s = fmaf(h[i], wv[i], s);
    out[t] = s;
}

extern "C" void kernel_launch(void* const* d_in, const int* in_sizes, int n_in,
                              void* d_out, int out_size, void* d_ws, size_t ws_size,
                              hipStream_t stream) {
    const float* x     = (const float*)d_in[0];
    const float* w     = (const float*)d_in[1];
    const float* q     = (const float*)d_in[2];
    const float* fc1_w = (const float*)d_in[3];
    const float* fc1_b = (const float*)d_in[4];
    const float* fc2_w = (const float*)d_in[5];
    const float* fc2_b = (const float*)d_in[6];
    float* out = (float*)d_out;

    float* Y = (float*)d_ws;             // [16, 7744] pooled dendrite output
    float* H = Y + BATCH * FIN;          // [16, 128]  fc1 activations

    dendrite_pool_kernel<<<BATCH * SP, 352, 0, stream>>>(x, w, q, Y);
    fc1_wmma_kernel<<<8, 32, 0, stream>>>(Y, fc1_w, fc1_b, H);
    fc2_kernel<<<1, 160, 0, stream>>>(H, fc2_w, fc2_b, out);
}